// nnlsConeAlignedCosine_23536420782164
// MI455X (gfx1250) — compile-verified
//
#include <hip/hip_runtime.h>
#include <math.h>

// Problem constants (match reference)
#define B_N 512
#define M_N 128
#define D_N 512
#define POW_ITERS 30
#define N_ITER 400

// Tiling
#define KC 64            // K-chunk staged in LDS
#define KC_PAD 66        // even stride: 8B-aligned frags, rows bank-staggered by 2
#define G_PAD 132        // multiple of 4: float4-aligned rows, bank-staggered by 4

// LDS layout (in floats)
#define OFF_CTRS 0
#define SZ_CTRS (M_N * KC_PAD)            // 8448
#define OFF_G    (OFF_CTRS + SZ_CTRS)
#define SZ_G     (M_N * G_PAD)            // 16896
#define OFF_B    (OFF_G + SZ_G)           // 25344 (mult of 4)
#define OFF_MASK (OFF_B + M_N)            // 25472
#define OFF_V    (OFF_MASK + M_N)         // 25600
#define OFF_LAM  (OFF_V + M_N)            // 25728
#define OFF_Y    (OFF_LAM + M_N)          // 25856
#define OFF_RED  (OFF_Y + M_N)            // 25984
#define SZ_RED   256
#define SMEM_FLOATS (OFF_RED + SZ_RED)    // 26240 floats = 104960 bytes

typedef float v2f __attribute__((ext_vector_type(2)));
typedef float v8f __attribute__((ext_vector_type(8)));

__device__ __forceinline__ float block_reduce_sum(float v, float* red) {
    const int t = threadIdx.x;
    red[t] = v;
    __syncthreads();
    #pragma unroll
    for (int s = 128; s > 0; s >>= 1) {
        if (t < s) red[t] += red[t + s];
        __syncthreads();
    }
    float r = red[0];
    __syncthreads();   // protect red[] for reuse
    return r;
}

// Vectorized LDS matvec row-dot: row must be float4-aligned (G_PAD % 4 == 0).
__device__ __forceinline__ float row_dot_f4(const float* __restrict__ grow,
                                            const float* __restrict__ vec) {
    const float4* __restrict__ g4 = (const float4*)grow;
    const float4* __restrict__ v4 = (const float4*)vec;
    float acc = 0.0f;
    #pragma unroll 8
    for (int j = 0; j < M_N / 4; ++j) {
        float4 g = g4[j];
        float4 v = v4[j];
        acc += g.x * v.x + g.y * v.y + g.z * v.z + g.w * v.w;
    }
    return acc;
}

// One workgroup (256 threads = 8 waves) per batch element.
__global__ void nnls_cone_cos_kernel(const float* __restrict__ pred_cost,
                                     const float* __restrict__ tight_ctrs,
                                     float* __restrict__ ws_neg_cos) {
    extern __shared__ float sm[];
    const int batch = blockIdx.x;
    const int t     = threadIdx.x;
    const int wave  = t >> 5;
    const int lane  = t & 31;

    const float* __restrict__ ctr_g  = tight_ctrs + (size_t)batch * M_N * D_N;
    const float* __restrict__ pred_g = pred_cost  + (size_t)batch * D_N;

    // ---------------- Phase 1: row mask + b = mask * (C @ (-p)) ----------------
    if (t < M_N) {
        const float4* row = (const float4*)(ctr_g + (size_t)t * D_N);
        const float4* pr4 = (const float4*)pred_g;
        float sa = 0.f, sb = 0.f;
        for (int d = 0; d < D_N / 4; ++d) {
            float4 a = row[d];
            float4 p = pr4[d];
            sa += fabsf(a.x) + fabsf(a.y) + fabsf(a.z) + fabsf(a.w);
            sb -= a.x * p.x + a.y * p.y + a.z * p.z + a.w * p.w;
        }
        float mk = (sa > 1e-7f) ? 1.0f : 0.0f;
        sm[OFF_MASK + t] = mk;
        sm[OFF_B + t]    = mk * sb;
    }

    // ---------------- Phase 2: G = C @ C^T via V_WMMA_F32_16X16X4_F32 ----------
    // Wave w computes output tile row w: tiles (w, 0..7), each 16x16.
    v8f acc[8];
    #pragma unroll
    for (int tn = 0; tn < 8; ++tn)
        #pragma unroll
        for (int j = 0; j < 8; ++j) acc[tn][j] = 0.0f;

    const int fr    = lane & 15;          // row within 16-tile (A: M, B: N)
    const int khalf = (lane >> 4) << 1;   // lane<16 -> K{0,1}; lane>=16 -> K{2,3}

    for (int kc = 0; kc < D_N; kc += KC) {
        __syncthreads();   // previous chunk fully consumed
        // Stage C[0:128][kc:kc+64] into LDS (float4 coalesced loads)
        for (int f = t; f < (M_N * KC) / 4; f += 256) {
            int m  = f >> 4;           // 16 float4 per row
            int k4 = (f & 15) << 2;
            const float4 v4 = *(const float4*)(ctr_g + (size_t)m * D_N + kc + k4);
            float* dst = &sm[OFF_CTRS + m * KC_PAD + k4];
            dst[0] = v4.x; dst[1] = v4.y; dst[2] = v4.z; dst[3] = v4.w;
        }
        __syncthreads();

        const float* aRow = &sm[OFF_CTRS + (wave * 16 + fr) * KC_PAD + khalf];
        #pragma unroll 4
        for (int k = 0; k < KC; k += 4) {
            // Issue ALL fragment loads first so WMMAs can use partial dscnt
            // waits and overlap DS traffic with matrix issue.
            v2f a = *(const v2f*)(aRow + k);
            v2f bf[8];
            #pragma unroll
            for (int tn = 0; tn < 8; ++tn)
                bf[tn] = *(const v2f*)(&sm[OFF_CTRS + (tn * 16 + fr) * KC_PAD + khalf + k]);
            #pragma unroll
            for (int tn = 0; tn < 8; ++tn)
                acc[tn] = __builtin_amdgcn_wmma_f32_16x16x4_f32(
                    false, a, false, bf[tn], (short)0, acc[tn], false, false);
        }
    }
    __syncthreads();

    // Write accumulators to G in LDS, applying mask outer product.
    // C/D layout: VGPR r -> M = r + (lane<16 ? 0 : 8), N = lane & 15.
    {
        const int mo = (lane >> 4) << 3;   // 0 or 8
        #pragma unroll
        for (int tn = 0; tn < 8; ++tn) {
            const int n  = tn * 16 + (lane & 15);
            const float mn = sm[OFF_MASK + n];
            #pragma unroll
            for (int r = 0; r < 8; ++r) {
                const int m = wave * 16 + r + mo;
                sm[OFF_G + m * G_PAD + n] = acc[tn][r] * sm[OFF_MASK + m] * mn;
            }
        }
    }
    __syncthreads();

    // ---------------- Phase 3: power iteration -> L = ||G||_2 ------------------
    if (t < M_N) sm[OFF_V + t] = 1.0f;
    __syncthreads();

    float L = 0.0f;
    for (int it = 0; it <= POW_ITERS; ++it) {
        float gv = 0.0f;
        if (t < M_N)
            gv = row_dot_f4(&sm[OFF_G + t * G_PAD], &sm[OFF_V]);
        float nrm = sqrtf(block_reduce_sum((t < M_N) ? gv * gv : 0.0f, &sm[OFF_RED]));
        if (it == POW_ITERS) { L = nrm; break; }
        if (t < M_N) sm[OFF_V + t] = gv / (nrm + 1e-12f);
        __syncthreads();
    }
    const float step = 1.0f / fmaxf(L, 1e-12f);

    // ---------------- Phase 4: FISTA with nonnegativity projection -------------
    if (t < M_N) { sm[OFF_LAM + t] = 0.0f; sm[OFF_Y + t] = 0.0f; }
    __syncthreads();

    float tk = 1.0f;
    for (int it = 0; it < N_ITER; ++it) {
        const float tnew = 0.5f * (1.0f + sqrtf(1.0f + 4.0f * tk * tk));
        float ln = 0.0f, yn = 0.0f;
        if (t < M_N) {
            float g = row_dot_f4(&sm[OFF_G + t * G_PAD], &sm[OFF_Y]) - sm[OFF_B + t];
            const float lamold = sm[OFF_LAM + t];
            ln = fmaxf(sm[OFF_Y + t] - step * g, 0.0f);
            yn = ln + ((tk - 1.0f) / tnew) * (ln - lamold);
        }
        __syncthreads();   // everyone done reading y/lam
        if (t < M_N) { sm[OFF_LAM + t] = ln; sm[OFF_Y + t] = yn; }
        tk = tnew;
        __syncthreads();
    }

    // ---------------- Phase 5: proj = lam^T C, cosine loss ---------------------
    float spp = 0.f, sqq = 0.f, sp2 = 0.f;
    for (int d = t; d < D_N; d += 256) {
        float pr = 0.0f;
        for (int m = 0; m < M_N; ++m)
            pr += sm[OFF_LAM + m] * ctr_g[(size_t)m * D_N + d];
        const float pd = -pred_g[d];
        spp += pd * pr;
        sqq += pr * pr;
        sp2 += pd * pd;
    }
    spp = block_reduce_sum(spp, &sm[OFF_RED]);
    sqq = block_reduce_sum(sqq, &sm[OFF_RED]);
    sp2 = block_reduce_sum(sp2, &sm[OFF_RED]);

    if (t == 0) {
        const float qv  = sqrtf(sqq);
        const float inv = 1.0f / (qv + 1e-12f);       // proj normalization
        const float qn  = fmaxf(qv * inv, 1e-8f);     // ||proj_normalized||
        const float pn  = fmaxf(sqrtf(sp2), 1e-8f);   // ||p||
        const float cosv = (spp * inv) / (pn * qn);
        ws_neg_cos[batch] = -cosv;
    }
}

__global__ void mean_reduce_kernel(const float* __restrict__ ws, float* __restrict__ out) {
    __shared__ float red[256];
    const int t = threadIdx.x;
    red[t] = ws[t] + ws[t + 256];
    __syncthreads();
    #pragma unroll
    for (int s = 128; s > 0; s >>= 1) {
        if (t < s) red[t] += red[t + s];
        __syncthreads();
    }
    if (t == 0) out[0] = red[0] * (1.0f / (float)B_N);
}

extern "C" void kernel_launch(void* const* d_in, const int* in_sizes, int n_in,
                              void* d_out, int out_size, void* d_ws, size_t ws_size,
                              hipStream_t stream) {
    const float* pred = (const float*)d_in[0];   // (B, D) f32
    const float* ctrs = (const float*)d_in[1];   // (B, M, D) f32
    float* out = (float*)d_out;                  // scalar f32
    float* ws  = (float*)d_ws;                   // >= 512 floats

    const size_t smem_bytes = (size_t)SMEM_FLOATS * sizeof(float);
    nnls_cone_cos_kernel<<<B_N, 256, smem_bytes, stream>>>(pred, ctrs, ws);
    mean_reduce_kernel<<<1, 256, 0, stream>>>(ws, out);
}